// Voxelization_80719615361768
// MI455X (gfx1250) — compile-verified
//
#include <hip/hip_runtime.h>

namespace {
constexpr int R  = 32;
constexpr int R3 = R * R * R;     // 32768 voxels
constexpr int C  = 64;            // channels
constexpr int B  = 8;             // batch
constexpr int N  = 65536;         // points per batch
}

// One thread per point. Within a wave, lanes hold 32 consecutive n, so each
// channel-iteration load features[b, c, n0..n0+31] is one coalesced 128B line.
// Features are streamed once -> non-temporal loads (keep 192MB L2 free for the
// 64MB vox accumulator, which atomics hammer). Atomics are relaxed agent-scope
// no-return f32 adds -> native global_atomic_add_f32 at L2 (STOREcnt tracked).
__global__ __launch_bounds__(256) void voxel_scatter(
    const float* __restrict__ feats,   // [B, C, N]
    const float* __restrict__ coords,  // [B, 3, N]
    float* __restrict__ vox,           // [B, C, R3]  (pre-zeroed)
    float* __restrict__ ncoords,       // [B, 3, N]
    float* __restrict__ cnt)           // [B, R3]     (pre-zeroed)
{
    const int t = blockIdx.x * blockDim.x + threadIdx.x;   // 0 .. B*N-1
    const int b = t >> 16;            // N == 65536
    const int n = t & (N - 1);

    const size_t cbase = (size_t)b * 3 * N + n;
    const float x = coords[cbase + 0 * (size_t)N];
    const float y = coords[cbase + 1 * (size_t)N];
    const float z = coords[cbase + 2 * (size_t)N];

    const float fx = fminf(fmaxf(x * (float)R, 0.0f), (float)(R - 1));
    const float fy = fminf(fmaxf(y * (float)R, 0.0f), (float)(R - 1));
    const float fz = fminf(fmaxf(z * (float)R, 0.0f), (float)(R - 1));

    // norm_coords: write-once -> non-temporal stores
    __builtin_nontemporal_store(fx, ncoords + cbase + 0 * (size_t)N);
    __builtin_nontemporal_store(fy, ncoords + cbase + 1 * (size_t)N);
    __builtin_nontemporal_store(fz, ncoords + cbase + 2 * (size_t)N);

    // rintf == round-half-to-even, matching jnp.round
    const int vx = (int)rintf(fx);
    const int vy = (int)rintf(fy);
    const int vz = (int)rintf(fz);
    const int idx = (vx * R + vy) * R + vz;

    (void)__hip_atomic_fetch_add(&cnt[b * R3 + idx], 1.0f,
                                 __ATOMIC_RELAXED, __HIP_MEMORY_SCOPE_AGENT);

    const float* __restrict__ src = feats + (size_t)b * C * N + n;
    float*       __restrict__ dst = vox   + (size_t)b * C * R3 + idx;

#pragma unroll 16
    for (int c = 0; c < C; ++c) {
        const float v = __builtin_nontemporal_load(src + (size_t)c * N);
        (void)__hip_atomic_fetch_add(dst + (size_t)c * R3, v,
                                     __ATOMIC_RELAXED, __HIP_MEMORY_SCOPE_AGENT);
    }
}

// In-place divide of the vox accumulator by max(count, 1), float4-vectorized.
// R3 is a multiple of 4, so the 4 elements share one (b, voxel-run).
__global__ __launch_bounds__(256) void voxel_finalize(
    float* __restrict__ vox, const float* __restrict__ cnt)
{
    const int t = blockIdx.x * blockDim.x + threadIdx.x;   // 0 .. B*C*R3/4-1
    const size_t i = (size_t)t * 4;
    const int b = (int)(i >> 21);          // / (C*R3) == / 2097152
    const int v = (int)(i & (size_t)(R3 - 1));
    const float* cb = cnt + (size_t)b * R3 + v;

    float4 s = *reinterpret_cast<float4*>(vox + i);
    s.x /= fmaxf(cb[0], 1.0f);
    s.y /= fmaxf(cb[1], 1.0f);
    s.z /= fmaxf(cb[2], 1.0f);
    s.w /= fmaxf(cb[3], 1.0f);
    *reinterpret_cast<float4*>(vox + i) = s;
}

extern "C" void kernel_launch(void* const* d_in, const int* in_sizes, int n_in,
                              void* d_out, int out_size, void* d_ws, size_t ws_size,
                              hipStream_t stream) {
    const float* feats  = (const float*)d_in[0];   // [B, C, N] f32
    const float* coords = (const float*)d_in[1];   // [B, 3, N] f32
    // d_in[2] = resolution (int32 scalar) == 32, compiled in as R.

    float* vox     = (float*)d_out;                        // B*C*R3 floats
    float* ncoords = (float*)d_out + (size_t)B * C * R3;   // B*3*N floats
    float* cnt     = (float*)d_ws;                         // B*R3 floats (1 MB)

    // Zero accumulators every call (harness poisons once; we must be
    // deterministic across graph replays). Memset nodes are capture-safe.
    hipMemsetAsync(vox, 0, (size_t)B * C * R3 * sizeof(float), stream);
    hipMemsetAsync(cnt, 0, (size_t)B * R3 * sizeof(float), stream);

    voxel_scatter<<<(B * N) / 256, 256, 0, stream>>>(feats, coords, vox,
                                                     ncoords, cnt);
    voxel_finalize<<<(B * C * R3 / 4) / 256, 256, 0, stream>>>(vox, cnt);
}